// SinkhornMatcher_23897198035441
// MI455X (gfx1250) — compile-verified
//
#include <hip/hip_runtime.h>
#include <math.h>

typedef __attribute__((ext_vector_type(2))) float v2f;
typedef __attribute__((ext_vector_type(8))) float v8f;

#define BATCH   8
#define NN      2048
#define DD      256
#define NP      2049            // padded (with dustbin)
#define LOGN    7.6246189861593985f   // log(2048)
#define NCHUNKS 8
#define CHUNK   257             // ceil(2049/8)

// ---------------- row squared-norms: one wave per descriptor row -------------
__global__ void __launch_bounds__(256)
norms_kernel(const float* __restrict__ d1, const float* __restrict__ d2,
             float* __restrict__ norms) {
  int gw   = (int)((blockIdx.x * blockDim.x + threadIdx.x) >> 5);
  int lane = threadIdx.x & 31;
  if (gw >= 2 * BATCH * NN) return;
  const float* src = (gw < BATCH * NN) ? d1 : d2;
  int row = gw % (BATCH * NN);
  const float* p = src + (size_t)row * DD;
  float s = 0.f;
#pragma unroll
  for (int i = 0; i < DD / 32; ++i) { float x = p[lane + 32 * i]; s += x * x; }
#pragma unroll
  for (int off = 16; off; off >>= 1) s += __shfl_xor(s, off, 32);
  if (lane == 0) norms[gw] = s;
}

// ---------------- dustbin row/col fill + u/v zero-init -----------------------
__global__ void __launch_bounds__(256)
init_kernel(float* __restrict__ out, float* __restrict__ u, float* __restrict__ v) {
  int i = blockIdx.x * blockDim.x + threadIdx.x;
  if (i < BATCH * NP) {                           // dustbin row n = 2048
    int b = i / NP, m = i % NP;
    out[((size_t)b * NP + NN) * NP + m] = -1.0f;
  } else if (i < BATCH * NP + BATCH * NN) {       // dustbin col m = 2048
    int j = i - BATCH * NP;
    int b = j / NN, n = j % NN;
    out[((size_t)b * NP + n) * NP + NN] = -1.0f;
  } else if (i < BATCH * NP + BATCH * NN + 2 * BATCH * NP) {
    int j = i - (BATCH * NP + BATCH * NN);
    if (j < BATCH * NP) u[j] = 0.f; else v[j - BATCH * NP] = 0.f;
  }
}

// ---------------- WMMA fp32 GEMM + squared-L2 score --------------------------
// grid (32, 16, 8): x = 64-col tile, y = 128-row tile, z = batch.
// Each wave: 16 rows x 64 cols = 4 WMMA tiles, K swept 4 at a time.
__global__ void __launch_bounds__(256)
gemm_score_kernel(const float* __restrict__ d1, const float* __restrict__ d2,
                  const float* __restrict__ norms, float* __restrict__ out) {
  const int b     = blockIdx.z;
  const int wave  = threadIdx.x >> 5;
  const int lane  = threadIdx.x & 31;
  const int lhalf = lane >> 4;        // 0: K=k,k+1   1: K=k+2,k+3
  const int l16   = lane & 15;
  const int n0    = blockIdx.y * 128 + wave * 16;
  const int m0    = blockIdx.x * 64;

  const float* arow = d1 + ((size_t)b * NN + n0 + l16) * DD;  // A row (M = l16)
  const float* brow = d2 + ((size_t)b * NN + m0 + l16) * DD;  // B col (N = l16)

  v8f acc0 = {}, acc1 = {}, acc2 = {}, acc3 = {};
  for (int k = 0; k < DD; k += 4) {
    const int kk = k + 2 * lhalf;
    v2f a;  a.x  = arow[kk];              a.y  = arow[kk + 1];
    v2f b0; b0.x = brow[kk];              b0.y = brow[kk + 1];
    v2f b1; b1.x = brow[16 * DD + kk];    b1.y = brow[16 * DD + kk + 1];
    v2f b2; b2.x = brow[32 * DD + kk];    b2.y = brow[32 * DD + kk + 1];
    v2f b3; b3.x = brow[48 * DD + kk];    b3.y = brow[48 * DD + kk + 1];
    acc0 = __builtin_amdgcn_wmma_f32_16x16x4_f32(false, a, false, b0, (short)0, acc0, false, false);
    acc1 = __builtin_amdgcn_wmma_f32_16x16x4_f32(false, a, false, b1, (short)0, acc1, false, false);
    acc2 = __builtin_amdgcn_wmma_f32_16x16x4_f32(false, a, false, b2, (short)0, acc2, false, false);
    acc3 = __builtin_amdgcn_wmma_f32_16x16x4_f32(false, a, false, b3, (short)0, acc3, false, false);
  }

  const float* n1b = norms + b * NN;
  const float* n2b = norms + BATCH * NN + b * NN;
  v8f accs[4] = {acc0, acc1, acc2, acc3};
#pragma unroll
  for (int j = 0; j < 4; ++j) {
    int col  = m0 + 16 * j + l16;
    float cn = n2b[col];
#pragma unroll
    for (int r = 0; r < 8; ++r) {
      int row = n0 + 8 * lhalf + r;                 // C/D layout: M = r + 8*lhalf
      float c = n1b[row] + cn - 2.f * accs[j][r];
      c = fmaxf(c, 0.f);
      out[((size_t)b * NP + row) * NP + col] = -c;  // EPSILON = 1
    }
  }
}

// ---------------- u update: one wave per row, lanes stride m -----------------
__global__ void __launch_bounds__(256)
u_kernel(const float* __restrict__ S, const float* __restrict__ v,
         float* __restrict__ u) {
  int wave = (int)((blockIdx.x * blockDim.x + threadIdx.x) >> 5);
  int lane = threadIdx.x & 31;
  if (wave >= BATCH * NP) return;
  int b = wave / NP, n = wave % NP;
  const float* row = S + ((size_t)b * NP + n) * NP;
  const float* vb  = v + b * NP;
  float s = 0.f;
  for (int j = lane; j < NP; j += 32) s += __expf(row[j] + vb[j]);
#pragma unroll
  for (int off = 16; off; off >>= 1) s += __shfl_xor(s, off, 32);
  if (lane == 0) {
    float logmu = (n == NN) ? LOGN : 0.f;
    u[b * NP + n] = logmu - __logf(s);
  }
}

// ---------------- v update stage 1: coalesced column partial sums ------------
// grid (9, NCHUNKS, 8): x*256+t = column, y = row chunk, z = batch
__global__ void __launch_bounds__(256)
v_partial_kernel(const float* __restrict__ S, const float* __restrict__ u,
                 float* __restrict__ psum) {
  int m = blockIdx.x * 256 + threadIdx.x;
  int b = blockIdx.z;
  if (m >= NP) return;
  int nbeg = blockIdx.y * CHUNK;
  int nend = nbeg + CHUNK; if (nend > NP) nend = NP;
  const float* Sb = S + (size_t)b * NP * NP;
  const float* ub = u + b * NP;
  float s = 0.f;
  for (int n = nbeg; n < nend; ++n) {
    if (n + 16 < nend) __builtin_prefetch(&Sb[(size_t)(n + 16) * NP + m], 0, 3);
    s += __expf(Sb[(size_t)n * NP + m] + ub[n]);
  }
  psum[((size_t)b * NCHUNKS + blockIdx.y) * NP + m] = s;
}

// ---------------- v update stage 2: combine partials -------------------------
__global__ void __launch_bounds__(256)
v_combine_kernel(const float* __restrict__ psum, float* __restrict__ v) {
  int i = blockIdx.x * blockDim.x + threadIdx.x;
  if (i >= BATCH * NP) return;
  int b = i / NP, m = i % NP;
  float s = 0.f;
#pragma unroll
  for (int c = 0; c < NCHUNKS; ++c) s += psum[((size_t)b * NCHUNKS + c) * NP + m];
  float lognu = (m == NN) ? LOGN : 0.f;
  v[i] = lognu - __logf(s);
}

// ---------------- final: P = exp(S + u + v) in place -------------------------
// grid (ceil(NP*NP/256), 8)
__global__ void __launch_bounds__(256)
final_kernel(float* __restrict__ out, const float* __restrict__ u,
             const float* __restrict__ v) {
  int idx = blockIdx.x * 256 + threadIdx.x;
  if (idx >= NP * NP) return;
  int b = blockIdx.y;
  int n = idx / NP, m = idx % NP;
  size_t o = (size_t)b * NP * NP + idx;
  out[o] = __expf(out[o] + u[b * NP + n] + v[b * NP + m]);
}

extern "C" void kernel_launch(void* const* d_in, const int* in_sizes, int n_in,
                              void* d_out, int out_size, void* d_ws, size_t ws_size,
                              hipStream_t stream) {
  const float* d1 = (const float*)d_in[0];
  const float* d2 = (const float*)d_in[1];
  float* out = (float*)d_out;
  float* ws  = (float*)d_ws;

  float* norms = ws;                         // 2 * 8 * 2048     = 32768 floats
  float* u     = norms + 2 * BATCH * NN;     // 8 * 2049         = 16392
  float* v     = u + BATCH * NP;             // 8 * 2049         = 16392
  float* psum  = v + BATCH * NP;             // 8 * 8 * 2049     = 131136

  // norms: 32768 waves -> 4096 blocks of 256
  norms_kernel<<<4096, 256, 0, stream>>>(d1, d2, norms);

  // dustbin + u/v init: 65560 items
  init_kernel<<<(BATCH * NP + BATCH * NN + 2 * BATCH * NP + 255) / 256, 256, 0, stream>>>(out, u, v);

  // WMMA GEMM + score
  gemm_score_kernel<<<dim3(32, 16, BATCH), 256, 0, stream>>>(d1, d2, norms, out);

  for (int it = 0; it < 20; ++it) {
    u_kernel<<<(BATCH * NP * 32 + 255) / 256, 256, 0, stream>>>(out, v, u);
    v_partial_kernel<<<dim3((NP + 255) / 256, NCHUNKS, BATCH), 256, 0, stream>>>(out, u, psum);
    v_combine_kernel<<<(BATCH * NP + 255) / 256, 256, 0, stream>>>(psum, v);
  }

  final_kernel<<<dim3((NP * NP + 255) / 256, BATCH), 256, 0, stream>>>(out, u, v);
}